// MelSpec_25864293057316
// MI455X (gfx1250) — compile-verified
//
#include <hip/hip_runtime.h>
#include <hip/hip_bf16.h>
#include <math.h>

// ---------------- problem constants (from reference) ----------------
#define LSIG     240000
#define BATCH    32
#define NFFT     512
#define HOPSZ    160
#define WINLEN   320
#define WINPAD   96          // (512-320)/2
#define NMELS    80
#define NBINS    257
#define TFRM     1501        // L/HOP + 1
#define TPAD     1504        // padded to multiple of 8
#define TILE_T   16
#define NTILES   94          // ceil(1501/16)
#define YLEN     ((TILE_T-1)*HOPSZ + NFFT)   // 2912 staged samples per tile
#define LOG_EPSF 5.9604644775390625e-8f      // 2^-24
#define BLANKV   27.0f
#define PREEMPH  0.97f
#define STD_EPSF 1e-5f
#define TWO_PI_F 6.28318530717958647692f

typedef __attribute__((ext_vector_type(2))) float v2f;
typedef __attribute__((ext_vector_type(8))) float v8f;

// ====================================================================
// Kernel 1: fused preemph + reflect pad + window + FFT512 + power +
//           mel projection (V_WMMA_F32_16X16X4_F32) + log
// One workgroup (256 thr = 8 waves) per (batch, 16-frame tile).
// ====================================================================
__global__ __launch_bounds__(256) void melspec_fft_wmma(
    const float* __restrict__ x,     // [32][240000]
    const float* __restrict__ fb,    // [80][257]
    float* __restrict__ out)         // [32][80][1504]
{
    __shared__ float yl[YLEN];            // staged preemphasized signal
    __shared__ float fre[TILE_T * NFFT];  // FFT real,  per frame
    __shared__ float fim[TILE_T * NFFT];  // FFT imag,  per frame
    __shared__ float Pp[NBINS * TILE_T];  // power, K-major [k][t]

    const int tid  = threadIdx.x;
    const int b    = blockIdx.x / NTILES;
    const int tile = blockIdx.x % NTILES;
    const int t0   = tile * TILE_T;
    const float* xb = x + (size_t)b * LSIG;

    // ---- Phase 1: stage y = preemph(x) with reflect padding into LDS ----
    // y_pad index j = t0*HOP - 256 + s ; reflect(j): j<0 -> -j, j>=L -> 2(L-1)-j
    for (int s = tid; s < YLEN; s += 256) {
        int j = t0 * HOPSZ - (NFFT / 2) + s;
        int jr = j;
        if (jr < 0)      jr = -jr;
        if (jr >= LSIG)  jr = 2 * (LSIG - 1) - jr;
        float v = (jr == 0) ? xb[0] : (xb[jr] - PREEMPH * xb[jr - 1]);
        yl[s] = v;
        // warm L2/L0 for the frames the *next* tile of this batch will touch
        int jn = j + TILE_T * HOPSZ;
        if (jn < LSIG) __builtin_prefetch(xb + jn, 0, 2);  // global_prefetch_b8
    }
    __syncthreads();

    // ---- Phase 2: windowed, bit-reversed load of 16 frames ----
    for (int e = tid; e < TILE_T * NFFT; e += 256) {
        int f = e >> 9;          // frame in tile
        int i = e & (NFFT - 1);  // sample in frame
        float w = 0.0f;
        if (i >= WINPAD && i < WINPAD + WINLEN) {
            w = 0.5f - 0.5f * __cosf(TWO_PI_F * (float)(i - WINPAD) / (float)WINLEN);
        }
        float val = yl[f * HOPSZ + i] * w;
        int ir = (int)(__brev((unsigned)i) >> 23);  // 9-bit reverse
        fre[f * NFFT + ir] = val;
        fim[f * NFFT + ir] = 0.0f;
    }
    __syncthreads();

    // ---- Phase 3: 16 x radix-2 DIT FFT-512, all threads cooperate ----
    // 4096 butterflies / stage, 16 per thread, 9 stages.
    for (int st = 0; st < 9; ++st) {
        const int half = 1 << st;
        const int span = half << 1;
        for (int u = tid; u < TILE_T * (NFFT / 2); u += 256) {
            int f   = u >> 8;
            int idx = u & 255;
            int pos = idx & (half - 1);
            int grp = idx >> st;
            int i0  = f * NFFT + grp * span + pos;
            int i1  = i0 + half;
            float ang = -TWO_PI_F * (float)pos / (float)span;
            float sn, cs;
            __sincosf(ang, &sn, &cs);
            float r1 = fre[i1], m1 = fim[i1];
            float tr = cs * r1 - sn * m1;
            float ti = cs * m1 + sn * r1;
            float r0 = fre[i0], m0 = fim[i0];
            fre[i1] = r0 - tr; fim[i1] = m0 - ti;
            fre[i0] = r0 + tr; fim[i0] = m0 + ti;
        }
        __syncthreads();
    }

    // ---- Phase 4: power spectrum, K-major for B-fragment feeds ----
    for (int e = tid; e < TILE_T * NBINS; e += 256) {
        int f = e / NBINS;
        int k = e - f * NBINS;
        float re = fre[f * NFFT + k];
        float im = fim[f * NFFT + k];
        Pp[k * TILE_T + f] = re * re + im * im;
    }
    __syncthreads();

    // ---- Phase 5: mel projection on the matrix pipe ----
    // D[16 mel x 16 frames] += A[16x4] * B[4x16], f32 WMMA.
    // Main loop: 64 unchecked K-steps (k=0..255); epilogue handles k=256.
    const int wave = tid >> 5;
    const int lane = tid & 31;
    if (wave < NMELS / 16) {                 // waves 0..4 own M-tiles
        const int m0   = wave * 16;
        const int lh   = lane & 15;
        const bool lo  = (lane < 16);
        const int koff = lo ? 0 : 2;          // ISA A/B frag: lanes 16-31 hold K+2,K+3
        const float* frow = fb + (size_t)(m0 + lh) * NBINS;  // filterbank row
        v8f acc = {};
        for (int k0 = 0; k0 < 256; k0 += 4) { // all in-bounds: kA+1 <= 255
            const int kA = k0 + koff;
            v2f a, bfr;
            a.x   = frow[kA];
            a.y   = frow[kA + 1];
            bfr.x = Pp[kA * TILE_T + lh];
            bfr.y = Pp[(kA + 1) * TILE_T + lh];
            acc = __builtin_amdgcn_wmma_f32_16x16x4_f32(
                false, a, false, bfr, (short)0, acc, false, false);
        }
        { // epilogue K-step: k = 256 valid, 257..259 are zero pad
            v2f a, bfr;
            a.x   = lo ? frow[256]            : 0.0f;
            a.y   = 0.0f;
            bfr.x = lo ? Pp[256 * TILE_T + lh] : 0.0f;
            bfr.y = 0.0f;
            acc = __builtin_amdgcn_wmma_f32_16x16x4_f32(
                false, a, false, bfr, (short)0, acc, false, false);
        }
        // ---- Phase 6: log + store. D frag: VGPR r = rows m0+r / m0+r+8 ----
        const int t = t0 + lh;
        if (t < TFRM) {
            const int mhi = lo ? 0 : 8;
            float* orow = out + (size_t)b * NMELS * TPAD + (size_t)t;
            #pragma unroll
            for (int r = 0; r < 8; ++r) {
                int mrow = m0 + r + mhi;
                orow[(size_t)mrow * TPAD] = __logf(acc[r] + LOG_EPSF);
            }
        }
    }
}

// ====================================================================
// Kernel 2: per-(b,mel) row normalization (mean/std over valid frames),
// zero invalid tail, BLANK pad to 1504, emit seq_len.
// One block per row, in-place on out.
// ====================================================================
__global__ __launch_bounds__(256) void melspec_normalize(
    const int* __restrict__ xn,      // [32]
    float* __restrict__ out)         // [32][80][1504] (+ seq_len tail)
{
    __shared__ float s1[256];
    __shared__ float s2[256];
    __shared__ float sMean, sInv;

    const int tid = threadIdx.x;
    const int row = blockIdx.x;      // b*80 + m
    const int b   = row / NMELS;
    const int m   = row - b * NMELS;
    const int n   = xn[b] / HOPSZ + 1;   // seq_len, >= 1001
    float* rp = out + (size_t)row * TPAD;

    float sum = 0.0f, sq = 0.0f;
    for (int t = tid; t < n; t += 256) {
        float v = rp[t];
        sum += v;
        sq  += v * v;
    }
    s1[tid] = sum; s2[tid] = sq;
    __syncthreads();
    for (int off = 128; off > 0; off >>= 1) {
        if (tid < off) { s1[tid] += s1[tid + off]; s2[tid] += s2[tid + off]; }
        __syncthreads();
    }
    if (tid == 0) {
        float fn   = (float)n;
        float mean = s1[0] / fn;
        float var  = fmaxf(s2[0] - fn * mean * mean, 0.0f) / (fn - 1.0f);
        sMean = mean;
        sInv  = 1.0f / (sqrtf(var) + STD_EPSF);
        if (m == 0) {
            ((int*)(out + (size_t)BATCH * NMELS * TPAD))[b] = n;  // seq_len output
        }
    }
    __syncthreads();

    const float mean = sMean, inv = sInv;
    for (int t = tid; t < TPAD; t += 256) {
        float o;
        if (t < n)          o = (rp[t] - mean) * inv;
        else if (t < TFRM)  o = 0.0f;     // masked frames
        else                o = BLANKV;   // pad-to-8 fill
        rp[t] = o;
    }
}

// ====================================================================
extern "C" void kernel_launch(void* const* d_in, const int* in_sizes, int n_in,
                              void* d_out, int out_size, void* d_ws, size_t ws_size,
                              hipStream_t stream) {
    (void)in_sizes; (void)n_in; (void)out_size; (void)d_ws; (void)ws_size;
    const float* x  = (const float*)d_in[0];   // (32, 240000) f32
    const int*   xn = (const int*)d_in[1];     // (32,) int
    const float* fb = (const float*)d_in[2];   // (80, 257) f32
    float* out = (float*)d_out;                // (32, 80, 1504) f32 + seq_len tail

    melspec_fft_wmma<<<dim3(BATCH * NTILES), dim3(256), 0, stream>>>(x, fb, out);
    melspec_normalize<<<dim3(BATCH * NMELS), dim3(256), 0, stream>>>(xn, out);
}